// BibdLinear_44341242364406
// MI455X (gfx1250) — compile-verified
//
#include <hip/hip_runtime.h>

typedef __attribute__((ext_vector_type(2))) float v2f;
typedef __attribute__((ext_vector_type(8))) float v8f;

#define IN_F  4096
#define OUT_F 4096
#define BATCH 8192

#define BM 128
#define BN 128
#define BK 32
#define LDA 36    // A-tile LDS stride (floats): rows 16B-aligned, conflict-free b64 frag loads
#define S2  129   // B-tile LDS stride in float2 units (odd -> low-conflict transposed stores)

__launch_bounds__(256, 2)
__global__ void bibd_gemm_wmma_f32(const float* __restrict__ x,
                                   const float* __restrict__ w,
                                   const float* __restrict__ mk,
                                   float* __restrict__ out) {
    // Double-buffered tiles.
    // lA [buf][row*LDA + k]          : x tile, row 0..127, k 0..31
    // lB2[buf][(k>>1)*S2 + n]        : (W*M)^T tile as K-pairs {B[k][n], B[k+1][n]}
    __shared__ float lA [2][BM * LDA];
    __shared__ v2f   lB2[2][(BK / 2) * S2];

    const int tid  = threadIdx.x;          // 0..255 (8 waves of 32)
    const int wv   = tid >> 5;             // wave id 0..7
    const int lane = tid & 31;

    const int bm = blockIdx.y * BM;        // batch-row block
    const int bn = blockIdx.x * BN;        // out-feature block

    // global->LDS staging: each thread moves 4x float4 per matrix per K-step
    const int rbase = tid >> 3;            // 0..31
    const int c4    = (tid & 7) * 4;       // 0,4,...,28
    const int p0    = c4 >> 1;             // B pair-row index for k = c4, c4+1

    // WMMA fragment coordinates (ISA 7.12.2, wave32)
    const int mrow  = wv * 16 + (lane & 15);   // A: both half-waves map M = lane&15
    const int khalf = (lane >> 4) * 2;         // lower half K={0,1}, upper half K={2,3}
    const int ncol  = lane & 15;               // N within 16-wide tile

    v8f zero = {};
    v8f acc[8];
    #pragma unroll
    for (int t = 0; t < 8; ++t) acc[t] = zero;

    // ---- prologue: stage K-tile 0 into buffer 0 ----
    #pragma unroll
    for (int j = 0; j < 4; ++j) {
        const int row = rbase + 32 * j;
        const float4 va = *reinterpret_cast<const float4*>(
            &x[(size_t)(bm + row) * IN_F + c4]);
        *reinterpret_cast<float4*>(&lA[0][row * LDA + c4]) = va;

        const float4 vw = *reinterpret_cast<const float4*>(
            &w[(size_t)(bn + row) * IN_F + c4]);
        const float4 vm = *reinterpret_cast<const float4*>(
            &mk[(size_t)(bn + row) * IN_F + c4]);
        v2f pa; pa.x = vw.x * vm.x; pa.y = vw.y * vm.y;
        v2f pb; pb.x = vw.z * vm.z; pb.y = vw.w * vm.w;
        lB2[0][(p0 + 0) * S2 + row] = pa;
        lB2[0][(p0 + 1) * S2 + row] = pb;
    }
    __syncthreads();

    int cur = 0;
    for (int ks = 0; ks < IN_F; ks += BK) {
        const int nxt = cur ^ 1;
        const bool have_next = (ks + BK) < IN_F;

        // ---- issue next K-tile's global loads into registers (overlaps compute) ----
        float4 va[4], vw4[4], vm4[4];
        if (have_next) {
            #pragma unroll
            for (int j = 0; j < 4; ++j) {
                const int row = rbase + 32 * j;
                va [j] = *reinterpret_cast<const float4*>(
                    &x[(size_t)(bm + row) * IN_F + ks + BK + c4]);
                vw4[j] = *reinterpret_cast<const float4*>(
                    &w[(size_t)(bn + row) * IN_F + ks + BK + c4]);
                vm4[j] = *reinterpret_cast<const float4*>(
                    &mk[(size_t)(bn + row) * IN_F + ks + BK + c4]);
            }
        }

        // ---- compute current tile: 8 K micro-steps x 8 N-tiles of WMMA ----
        const float* __restrict__ lAc = lA [cur];
        const v2f*   __restrict__ lBc = lB2[cur];
        #pragma unroll
        for (int kk = 0; kk < BK; kk += 4) {
            const v2f a = *reinterpret_cast<const v2f*>(
                &lAc[mrow * LDA + kk + khalf]);               // one ds_load_b64
            const int bprow = (kk + khalf) >> 1;              // pair-row 0..15
            #pragma unroll
            for (int t = 0; t < 8; ++t) {
                const v2f b = lBc[bprow * S2 + t * 16 + ncol]; // one ds_load_b64
                acc[t] = __builtin_amdgcn_wmma_f32_16x16x4_f32(
                    /*neg_a=*/false, a, /*neg_b=*/false, b,
                    /*c_mod=*/(short)0, acc[t],
                    /*reuse_a=*/false, /*reuse_b=*/false);
            }
        }

        // ---- drain staged registers into the other LDS buffer ----
        if (have_next) {
            #pragma unroll
            for (int j = 0; j < 4; ++j) {
                const int row = rbase + 32 * j;
                *reinterpret_cast<float4*>(&lA[nxt][row * LDA + c4]) = va[j];
                v2f pa; pa.x = vw4[j].x * vm4[j].x; pa.y = vw4[j].y * vm4[j].y;
                v2f pb; pb.x = vw4[j].z * vm4[j].z; pb.y = vw4[j].w * vm4[j].w;
                lB2[nxt][(p0 + 0) * S2 + row] = pa;
                lB2[nxt][(p0 + 1) * S2 + row] = pb;
            }
        }
        __syncthreads();
        cur = nxt;
    }

    // ---- epilogue: C/D layout. VGPR r: lanes 0-15 -> M=r, lanes 16-31 -> M=r+8 ----
    const int orow0 = bm + wv * 16 + ((lane >> 4) << 3);
    const int ocol0 = bn + (lane & 15);
    #pragma unroll
    for (int t = 0; t < 8; ++t) {
        #pragma unroll
        for (int r = 0; r < 8; ++r) {
            out[(size_t)(orow0 + r) * OUT_F + ocol0 + t * 16] = acc[t][r];
        }
    }
}

extern "C" void kernel_launch(void* const* d_in, const int* in_sizes, int n_in,
                              void* d_out, int out_size, void* d_ws, size_t ws_size,
                              hipStream_t stream) {
    const float* x  = (const float*)d_in[0];   // [BATCH, IN_F]
    const float* w  = (const float*)d_in[1];   // [OUT_F, IN_F]
    const float* mk = (const float*)d_in[2];   // [OUT_F, IN_F]
    float* out = (float*)d_out;                // [BATCH, OUT_F]

    dim3 grid(OUT_F / BN, BATCH / BM);         // (32, 64)
    dim3 block(256);                           // 8 waves (wave32)
    bibd_gemm_wmma_f32<<<grid, block, 0, stream>>>(x, w, mk, out);
}